// GraphConvLayer_15659450761598
// MI455X (gfx1250) — compile-verified
//
#include <hip/hip_runtime.h>

typedef __attribute__((ext_vector_type(2))) float v2f;
typedef __attribute__((ext_vector_type(8))) float v8f;

#define IN_DIM 128
#define OUT_DIM 128
#define LDS_STRIDE 132  // padded row stride: conflict-free ds_load_b64 A-fragments
#define MTILES 4        // 4 x 16 = 64 output rows per block

// One block = 8 wave32. Block computes a 64x128 output tile:
//   Out[node0:node0+64, :] = X[node0:node0+64, :] @ B (+ bias broadcast over rows)
// Each wave owns a 16-column slice of N and 4 independent 16x16 accumulators
// (4 independent WMMA C-chains; one B fragment feeds 4 WMMAs per k-step).
// BT=true : B[k][n] = W[n*IN_DIM + k]   (w_self stored [OUT,IN]; x @ w_self^T)
// BT=false: B[k][n] = W[k*OUT_DIM + n]  (relation_weights stored [IN,OUT])
template <bool BT>
__global__ __launch_bounds__(256) void rgcn_gemm_kernel(
    const float* __restrict__ X, const float* __restrict__ W,
    const float* __restrict__ bias, float* __restrict__ Out, int n_nodes) {
  __shared__ float As[16 * MTILES * LDS_STRIDE];
  const int node0 = blockIdx.x * (16 * MTILES);
  const int t = threadIdx.x;

  // Cooperative stage of A tile (64 rows x 128 cols of x): 32 floats/thread.
#pragma unroll
  for (int i = 0; i < MTILES; ++i) {
    int row = (t >> 4) + i * 16;  // 0..63
    int col = (t & 15) * 8;       // 0..120
    int gr = node0 + row;
    if (gr >= n_nodes) gr = n_nodes - 1;  // clamp; stores are guarded below
    const float4* gp = (const float4*)(X + (size_t)gr * IN_DIM + col);
    float4 p0 = gp[0];
    float4 p1 = gp[1];
    *(float4*)&As[row * LDS_STRIDE + col] = p0;
    *(float4*)&As[row * LDS_STRIDE + col + 4] = p1;
  }
  __syncthreads();

  const int lane = t & 31;
  const int wave = t >> 5;            // 0..7 -> 16-column slice of N=128
  const int ncol0 = wave * 16;
  const int nm = lane & 15;           // A: M row ; B/C/D: N column
  const int khalf = (lane >> 4) * 2;  // f32 16x16x4 layout: lanes16-31 hold K+2

  v8f c[MTILES];
#pragma unroll
  for (int mt = 0; mt < MTILES; ++mt) c[mt] = (v8f){};

#pragma unroll
  for (int k0 = 0; k0 < IN_DIM; k0 += 4) {
    const int kk = k0 + khalf;
    // B fragment: lane holds B[kk][N=nm], B[kk+1][N=nm]; shared by 4 WMMAs.
    v2f b;
    if (BT) {
      b = *(const v2f*)(W + (size_t)(ncol0 + nm) * IN_DIM + kk);  // contiguous k
    } else {
      b.x = W[(size_t)kk * OUT_DIM + ncol0 + nm];
      b.y = W[(size_t)(kk + 1) * OUT_DIM + ncol0 + nm];
    }
#pragma unroll
    for (int mt = 0; mt < MTILES; ++mt) {
      // A fragment: lane holds X[M][kk], X[M][kk+1]  (ds_load_2addr_b64)
      v2f a = *(v2f*)&As[(mt * 16 + nm) * LDS_STRIDE + kk];
      // D = A(16x4,f32) * B(4x16,f32) + C — exact f32 matrix op; 4 indep chains
      c[mt] = __builtin_amdgcn_wmma_f32_16x16x4_f32(false, a, false, b,
                                                    (short)0, c[mt], false,
                                                    false);
    }
  }

  // Epilogue: bias is per-column; every accumulator of a lane shares N.
  const float bv = bias[ncol0 + nm];
  const int hi8 = (lane >> 4) * 8;  // C/D layout: lanes16-31 hold M = i+8
#pragma unroll
  for (int mt = 0; mt < MTILES; ++mt) {
#pragma unroll
    for (int i = 0; i < 8; ++i) {
      int row = node0 + mt * 16 + hi8 + i;
      if (row < n_nodes) Out[(size_t)row * OUT_DIM + ncol0 + nm] = c[mt][i] + bv;
    }
  }
}

// One wave per edge; lane covers 4 output dims (float4 gather + 4 f32 atomics).
__global__ __launch_bounds__(256) void rgcn_scatter_kernel(
    const int* __restrict__ srcI, const int* __restrict__ dstI,
    const int* __restrict__ etype, const float* __restrict__ XR,
    float* __restrict__ Acc, int rel, int n_edges) {
  int e = blockIdx.x * 8 + (threadIdx.x >> 5);
  if (e >= n_edges) return;
  if (etype[e] != rel) return;  // wave-uniform branch
  int lane = threadIdx.x & 31;
  int s = srcI[e];
  int d = dstI[e];
  float4 v = ((const float4*)(XR + (size_t)s * OUT_DIM))[lane];
  float* op = Acc + (size_t)d * OUT_DIM + lane * 4;
  atomicAdd(op + 0, v.x);
  atomicAdd(op + 1, v.y);
  atomicAdd(op + 2, v.z);
  atomicAdd(op + 3, v.w);
}

// One wave per row: in-place LayerNorm over 128 dims (4 per lane, shfl_xor tree).
__global__ __launch_bounds__(256) void rgcn_layernorm_kernel(
    float* __restrict__ data, const float* __restrict__ gamma,
    const float* __restrict__ beta, int n_nodes) {
  int row = blockIdx.x * 8 + (threadIdx.x >> 5);
  if (row >= n_nodes) return;
  int lane = threadIdx.x & 31;
  float4 v = ((float4*)(data + (size_t)row * OUT_DIM))[lane];
  float s = v.x + v.y + v.z + v.w;
#pragma unroll
  for (int off = 16; off > 0; off >>= 1) s += __shfl_xor(s, off, 32);
  float mean = s * (1.0f / OUT_DIM);
  float d0 = v.x - mean, d1 = v.y - mean, d2 = v.z - mean, d3 = v.w - mean;
  float q = d0 * d0 + d1 * d1 + d2 * d2 + d3 * d3;
#pragma unroll
  for (int off = 16; off > 0; off >>= 1) q += __shfl_xor(q, off, 32);
  float inv = rsqrtf(q * (1.0f / OUT_DIM) + 1e-5f);
  float4 g = ((const float4*)gamma)[lane];
  float4 b = ((const float4*)beta)[lane];
  float4 o;
  o.x = d0 * inv * g.x + b.x;
  o.y = d1 * inv * g.y + b.y;
  o.z = d2 * inv * g.z + b.z;
  o.w = d3 * inv * g.w + b.w;
  ((float4*)(data + (size_t)row * OUT_DIM))[lane] = o;
}

extern "C" void kernel_launch(void* const* d_in, const int* in_sizes, int n_in,
                              void* d_out, int out_size, void* d_ws,
                              size_t ws_size, hipStream_t stream) {
  const float* x = (const float*)d_in[0];
  const int* edge_index = (const int*)d_in[1];
  const int* edge_type = (const int*)d_in[2];
  const float* relw = (const float*)d_in[3];
  const float* relb = (const float*)d_in[4];
  const float* wself = (const float*)d_in[5];
  const float* bself = (const float*)d_in[6];
  const float* gamma = (const float*)d_in[7];
  const float* beta = (const float*)d_in[8];
  float* out = (float*)d_out;
  float* xr = (float*)d_ws;  // 50000*128*4 = 25.6 MB scratch, rewritten per relation

  const int n_edges = in_sizes[2];
  const int n_nodes = in_sizes[0] / IN_DIM;
  const int n_rel = in_sizes[4] / OUT_DIM;
  const int* srcI = edge_index;
  const int* dstI = edge_index + n_edges;

  const int rows_per_blk = 16 * MTILES;
  const int mtiles = (n_nodes + rows_per_blk - 1) / rows_per_blk;
  const int eblocks = (n_edges + 7) / 8;
  const int rblocks = (n_nodes + 7) / 8;
  dim3 blk(256);

  // 1) Self-loop linear straight into d_out (fully overwrites poison).
  rgcn_gemm_kernel<true><<<mtiles, blk, 0, stream>>>(x, wself, bself, out, n_nodes);

  // 2) Per relation: xr = x @ W[r] + bias[r], then scatter-add matching edges.
  for (int r = 0; r < n_rel; ++r) {
    rgcn_gemm_kernel<false><<<mtiles, blk, 0, stream>>>(
        x, relw + (size_t)r * IN_DIM * OUT_DIM, relb + (size_t)r * OUT_DIM, xr,
        n_nodes);
    rgcn_scatter_kernel<<<eblocks, blk, 0, stream>>>(srcI, dstI, edge_type, xr,
                                                     out, r, n_edges);
  }

  // 3) In-place LayerNorm.
  rgcn_layernorm_kernel<<<rblocks, blk, 0, stream>>>(out, gamma, beta, n_nodes);
}